// MemoryEfficientAttention_22110491640218
// MI455X (gfx1250) — compile-verified
//
#include <hip/hip_runtime.h>
#include <hip/hip_bf16.h>

typedef __attribute__((ext_vector_type(2))) float v2f;
typedef __attribute__((ext_vector_type(8))) float v8f;

#define DMODEL 1024
#define HEADS  16
#define DHEAD  64
#define BATCH  2

__device__ __forceinline__ v8f wmma4(v2f a, v2f b, v8f c) {
  // V_WMMA_F32_16X16X4_F32: D = A(16x4) x B(4x16) + C(16x16), fp32 throughout.
  return __builtin_amdgcn_wmma_f32_16x16x4_f32(false, a, false, b, (short)0, c, false, false);
}

// ---------------------------------------------------------------------------
// GEMM: C[M x N] = A[M x K] @ B[K x N] (+ bias).  One wave per 16x64 C-tile.
// A-frag (lane l, vgpr j): A[m = l%16][k = k0 + j + 2*(l/16)]
// B-frag (lane l, vgpr j): B[k = k0 + j + 2*(l/16)][n = n0 + t*16 + l%16]
// C/D   (lane l, vgpr r):  C[m = r + 8*(l/16)][n = n0 + t*16 + l%16]
// ---------------------------------------------------------------------------
__global__ __launch_bounds__(32) void gemm_f32(const float* __restrict__ A,
                                               const float* __restrict__ B,
                                               const float* __restrict__ bias,
                                               float* __restrict__ C,
                                               int M, int N, int K) {
  const int lane = threadIdx.x;
  const int half = lane >> 4;   // 0 or 1
  const int l16  = lane & 15;
  const int m0 = blockIdx.x * 16;
  const int n0 = blockIdx.y * 64;

  v8f acc[4] = {};
  const float* arow = A + (size_t)(m0 + l16) * K + 2 * half;

  for (int k0 = 0; k0 < K; k0 += 4) {
    v2f af;
    af.x = arow[k0];
    af.y = arow[k0 + 1];
    const float* b0 = B + (size_t)(k0 + 2 * half) * N + n0 + l16;
    const float* b1 = b0 + N;
#pragma unroll
    for (int t = 0; t < 4; ++t) {
      v2f bf;
      bf.x = b0[t * 16];
      bf.y = b1[t * 16];
      acc[t] = wmma4(af, bf, acc[t]);
    }
  }

#pragma unroll
  for (int t = 0; t < 4; ++t) {
    const float bv = bias ? bias[n0 + t * 16 + l16] : 0.0f;
#pragma unroll
    for (int r = 0; r < 8; ++r) {
      C[(size_t)(m0 + r + 8 * half) * N + n0 + t * 16 + l16] = acc[t][r] + bv;
    }
  }
}

// ---------------------------------------------------------------------------
// 2D RoPE applied in-place to Q and K, layout (b, n, heads, DHEAD).
// pair p < 16 uses row angle, p >= 16 uses col angle; inv_freq = theta^(-2i/32)
// ---------------------------------------------------------------------------
__global__ void rope2d(float* __restrict__ Q, float* __restrict__ K,
                       int nrows /* b*n */, int n, const int* __restrict__ wp) {
  const int pairs = DHEAD / 2;           // 32
  const int idx = blockIdx.x * blockDim.x + threadIdx.x;
  const int total = nrows * HEADS * pairs;
  if (idx >= total) return;

  const int p    = idx % pairs;
  int rest       = idx / pairs;
  const int head = rest % HEADS;
  const int rI   = rest / HEADS;         // 0..nrows-1 (batch-major rows)
  const int nidx = rI % n;
  const int ww   = *wp;
  const int row  = nidx / ww;
  const int col  = nidx % ww;

  const int hp = pairs / 2;              // 16
  float pos;
  int fi;
  if (p < hp) { pos = (float)row; fi = p; }
  else        { pos = (float)col; fi = p - hp; }

  const float inv_freq = exp2f(-((float)(2 * fi) / (float)pairs) * log2f(10000.0f));
  const float ang = pos * inv_freq;
  const float c = cosf(ang), s = sinf(ang);

  const size_t off = (size_t)rI * DMODEL + head * DHEAD + 2 * p;
  const float qr = Q[off], qi = Q[off + 1];
  Q[off]     = qr * c - qi * s;
  Q[off + 1] = qr * s + qi * c;
  const float kr = K[off], ki = K[off + 1];
  K[off]     = kr * c - ki * s;
  K[off + 1] = kr * s + ki * c;
}

// ---------------------------------------------------------------------------
// Flash attention: one wave per (batch, head, 16-row q tile). Online softmax.
// Q/K/V/O layout: (b, n, heads, DHEAD) i.e. stride DMODEL between tokens.
// ---------------------------------------------------------------------------
#define PSTRIDE 20  // padded LDS row stride (floats) -> conflict-free

__global__ __launch_bounds__(32) void flash_attn(const float* __restrict__ Q,
                                                 const float* __restrict__ K,
                                                 const float* __restrict__ V,
                                                 float* __restrict__ O,
                                                 int n) {
  __shared__ float pshare[16 * PSTRIDE];

  const int lane = threadIdx.x;
  const int half = lane >> 4;
  const int l16  = lane & 15;
  const int q0    = blockIdx.x * 16;
  const int head  = blockIdx.y;
  const int batch = blockIdx.z;
  const size_t base = (size_t)batch * n * DMODEL + head * DHEAD;

  // Preload Q tile as 16 A-fragments (covers k = 0..63 in blocks of 4)
  v2f qa[16];
  {
    const float* qrow = Q + base + (size_t)(q0 + l16) * DMODEL + 2 * half;
#pragma unroll
    for (int kb = 0; kb < 16; ++kb) {
      qa[kb].x = qrow[kb * 4];
      qa[kb].y = qrow[kb * 4 + 1];
    }
  }

  v8f o0 = {}, o1 = {}, o2 = {}, o3 = {};
  float mrow[8], lrow[8];
#pragma unroll
  for (int r = 0; r < 8; ++r) { mrow[r] = -1e30f; lrow[r] = 0.0f; }

  for (int kv0 = 0; kv0 < n; kv0 += 16) {
    // ---- S = Q x K^T  (B-frag of K^T == A-style load from row-major K) ----
    v8f s = {};
    const float* krow = K + base + (size_t)(kv0 + l16) * DMODEL + 2 * half;
#pragma unroll
    for (int kb = 0; kb < 16; ++kb) {
      v2f bf;
      bf.x = krow[kb * 4];
      bf.y = krow[kb * 4 + 1];
      s = wmma4(qa[kb], bf, s);
    }
#pragma unroll
    for (int r = 0; r < 8; ++r) s[r] *= 0.125f;  // 1/sqrt(64)

    // ---- online softmax: row max / exp / row sum (reduce within 16-lane halves)
    v8f pexp;
#pragma unroll
    for (int r = 0; r < 8; ++r) {
      float v = s[r];
      v = fmaxf(v, __shfl_xor(v, 1));
      v = fmaxf(v, __shfl_xor(v, 2));
      v = fmaxf(v, __shfl_xor(v, 4));
      v = fmaxf(v, __shfl_xor(v, 8));
      const float mnew  = fmaxf(mrow[r], v);
      const float alpha = expf(mrow[r] - mnew);
      mrow[r] = mnew;
      const float p = expf(s[r] - mnew);
      pexp[r] = p;
      float sum = p;
      sum += __shfl_xor(sum, 1);
      sum += __shfl_xor(sum, 2);
      sum += __shfl_xor(sum, 4);
      sum += __shfl_xor(sum, 8);
      lrow[r] = lrow[r] * alpha + sum;
      o0[r] *= alpha; o1[r] *= alpha; o2[r] *= alpha; o3[r] *= alpha;
    }

    // ---- re-fragment P via LDS: C/D layout -> A layout ----
#pragma unroll
    for (int r = 0; r < 8; ++r)
      pshare[(r + 8 * half) * PSTRIDE + l16] = pexp[r];
    __syncthreads();

    // ---- O += P x V ----
#pragma unroll
    for (int kb = 0; kb < 4; ++kb) {
      v2f pa;
      pa.x = pshare[l16 * PSTRIDE + kb * 4 + 2 * half];
      pa.y = pshare[l16 * PSTRIDE + kb * 4 + 2 * half + 1];
      const float* vr0 = V + base + (size_t)(kv0 + kb * 4 + 2 * half) * DMODEL + l16;
      const float* vr1 = vr0 + DMODEL;
      v2f b0, b1, b2, b3;
      b0.x = vr0[0];  b0.y = vr1[0];
      b1.x = vr0[16]; b1.y = vr1[16];
      b2.x = vr0[32]; b2.y = vr1[32];
      b3.x = vr0[48]; b3.y = vr1[48];
      o0 = wmma4(pa, b0, o0);
      o1 = wmma4(pa, b1, o1);
      o2 = wmma4(pa, b2, o2);
      o3 = wmma4(pa, b3, o3);
    }
    __syncthreads();
  }

  // ---- epilogue: O /= l, store ----
#pragma unroll
  for (int r = 0; r < 8; ++r) {
    const float il = 1.0f / lrow[r];
    const size_t off = base + (size_t)(q0 + r + 8 * half) * DMODEL + l16;
    O[off + 0]  = o0[r] * il;
    O[off + 16] = o1[r] * il;
    O[off + 32] = o2[r] * il;
    O[off + 48] = o3[r] * il;
  }
}

// ---------------------------------------------------------------------------
extern "C" void kernel_launch(void* const* d_in, const int* in_sizes, int n_in,
                              void* d_out, int out_size, void* d_ws, size_t ws_size,
                              hipStream_t stream) {
  const float* x  = (const float*)d_in[0];
  const float* Wq = (const float*)d_in[1];
  const float* Wk = (const float*)d_in[2];
  const float* Wv = (const float*)d_in[3];
  const float* Wo = (const float*)d_in[4];
  const float* bo = (const float*)d_in[5];
  const int*   wp = (const int*)d_in[7];  // w (device scalar)

  const int M = in_sizes[0] / DMODEL;     // b * n = 4608
  const int n = M / BATCH;                // tokens per batch = 2304

  const size_t MD = (size_t)M * DMODEL;
  float* Qb = (float*)d_ws;
  float* Kb = Qb + MD;
  float* Vb = Kb + MD;
  float* Ob = Vb + MD;

  dim3 gg(M / 16, DMODEL / 64);

  // QKV projections (fp32 WMMA GEMMs)
  gemm_f32<<<gg, 32, 0, stream>>>(x, Wq, nullptr, Qb, M, DMODEL, DMODEL);
  gemm_f32<<<gg, 32, 0, stream>>>(x, Wk, nullptr, Kb, M, DMODEL, DMODEL);
  gemm_f32<<<gg, 32, 0, stream>>>(x, Wv, nullptr, Vb, M, DMODEL, DMODEL);

  // 2D RoPE on Q and K (in place)
  {
    const int total = M * HEADS * (DHEAD / 2);
    rope2d<<<(total + 255) / 256, 256, 0, stream>>>(Qb, Kb, M, n, wp);
  }

  // Flash attention
  flash_attn<<<dim3(n / 16, HEADS, BATCH), 32, 0, stream>>>(Qb, Kb, Vb, Ob, n);

  // Output projection + bias
  gemm_f32<<<gg, 32, 0, stream>>>(Ob, Wo, bo, (float*)d_out, M, DMODEL, DMODEL);
}